// WeightedSumSessEmbedding_14362370638521
// MI455X (gfx1250) — compile-verified
//
#include <hip/hip_runtime.h>

typedef float v2f __attribute__((ext_vector_type(2)));
typedef float v8f __attribute__((ext_vector_type(8)));

#define EMB_DIM 64
#define TILE_M 16
#define WAVES_PER_BLOCK 4
#define K_CHUNK 4
#define BLOCK_THREADS (WAVES_PER_BLOCK * 32)

__device__ __forceinline__ int lower_bound_i32(const int* __restrict__ arr, int n, int key) {
  int lo = 0, hi = n;
  while (lo < hi) {
    int mid = (lo + hi) >> 1;
    if (arr[mid] < key) lo = mid + 1; else hi = mid;
  }
  return lo;
}

// One workgroup (4 x wave32) per tile of 16 consecutive sessions.
// Per K-chunk of 4 nnz:  D[16x64] += A[16x4] (weighted selector) x B[4x64] (gathered rows)
// via 4x V_WMMA_F32_16X16X4_F32 (one per 16-col N-tile), all in f32.
__global__ __launch_bounds__(BLOCK_THREADS)
void sess_emb_wmma_kernel(const int* __restrict__ row_idx,
                          const int* __restrict__ col_idx,
                          const float* __restrict__ data,
                          const float* __restrict__ emb,
                          float* __restrict__ out,
                          int nnz) {
  __shared__ float tile_lds[TILE_M * EMB_DIM];

  const int tid  = threadIdx.x;
  const int wave = tid >> 5;      // wave32: 4 waves per block
  const int lane = tid & 31;
  const int h    = lane >> 4;     // lane half (0: lanes 0-15, 1: lanes 16-31)
  const int m    = lane & 15;     // A row / B,C column within tile
  const int base_row = blockIdx.x * TILE_M;
  const int my_row   = base_row + m;

  // zero the reduction buffer
  for (int i = tid; i < TILE_M * EMB_DIM; i += BLOCK_THREADS) tile_lds[i] = 0.0f;

  // nnz range owned by this 16-session tile (row_idx sorted ascending)
  const int start = lower_bound_i32(row_idx, nnz, base_row);
  const int end   = lower_bound_i32(row_idx, nnz, base_row + TILE_M);

  v8f acc0 = {}, acc1 = {}, acc2 = {}, acc3 = {};

  // each wave takes every WAVES_PER_BLOCK-th chunk of K_CHUNK nnz
  int p = start + K_CHUNK * wave;

  // ---- fast path: full chunks, no clamping, all loads unconditional ----
  for (; p + K_CHUNK <= end; p += K_CHUNK * WAVES_PER_BLOCK) {
    // per-lane K indices per the 16x4 f32 A layout: VGPR0 <-> K=2h, VGPR1 <-> K=2h+1
    const int k0 = p + 2 * h;
    const int k1 = k0 + 1;

    const int   r0 = row_idx[k0];
    const int   r1 = row_idx[k1];
    const float w0 = data[k0];
    const float w1 = data[k1];

    // A = weighted selector: A[m][k] = data[k] iff row_idx[k] == base_row + m
    v2f a;
    a.x = (r0 == my_row) ? w0 : 0.0f;
    a.y = (r1 == my_row) ? w1 : 0.0f;

    // B = gathered embedding rows; per half-wave the 16 lanes read 64B contiguous
    const float* e0 = emb + (long long)col_idx[k0] * EMB_DIM;
    const float* e1 = emb + (long long)col_idx[k1] * EMB_DIM;

    v2f b0, b1, b2, b3;
    b0.x = e0[m];      b0.y = e1[m];
    b1.x = e0[m + 16]; b1.y = e1[m + 16];
    b2.x = e0[m + 32]; b2.y = e1[m + 32];
    b3.x = e0[m + 48]; b3.y = e1[m + 48];

    acc0 = __builtin_amdgcn_wmma_f32_16x16x4_f32(false, a, false, b0, (short)0, acc0, false, false);
    acc1 = __builtin_amdgcn_wmma_f32_16x16x4_f32(false, a, false, b1, (short)0, acc1, false, false);
    acc2 = __builtin_amdgcn_wmma_f32_16x16x4_f32(false, a, false, b2, (short)0, acc2, false, false);
    acc3 = __builtin_amdgcn_wmma_f32_16x16x4_f32(false, a, false, b3, (short)0, acc3, false, false);
  }

  // ---- tail: at most one partial chunk per wave; clamp addresses, zero weights ----
  if (p < end) {
    const int k0  = p + 2 * h;
    const int k1  = k0 + 1;
    const int kk0 = (k0 < end) ? k0 : (end - 1);   // valid address; weight zeroed below
    const int kk1 = (k1 < end) ? k1 : (end - 1);

    // unconditional loads at clamped (valid) addresses -> no exec-mask branches
    const int   r0 = row_idx[kk0];
    const int   r1 = row_idx[kk1];
    float       w0 = data[kk0];
    float       w1 = data[kk1];
    if (k0 >= end) w0 = 0.0f;
    if (k1 >= end) w1 = 0.0f;

    v2f a;
    a.x = (r0 == my_row) ? w0 : 0.0f;
    a.y = (r1 == my_row) ? w1 : 0.0f;

    const float* e0 = emb + (long long)col_idx[kk0] * EMB_DIM;
    const float* e1 = emb + (long long)col_idx[kk1] * EMB_DIM;

    v2f b0, b1, b2, b3;
    b0.x = e0[m];      b0.y = e1[m];
    b1.x = e0[m + 16]; b1.y = e1[m + 16];
    b2.x = e0[m + 32]; b2.y = e1[m + 32];
    b3.x = e0[m + 48]; b3.y = e1[m + 48];

    acc0 = __builtin_amdgcn_wmma_f32_16x16x4_f32(false, a, false, b0, (short)0, acc0, false, false);
    acc1 = __builtin_amdgcn_wmma_f32_16x16x4_f32(false, a, false, b1, (short)0, acc1, false, false);
    acc2 = __builtin_amdgcn_wmma_f32_16x16x4_f32(false, a, false, b2, (short)0, acc2, false, false);
    acc3 = __builtin_amdgcn_wmma_f32_16x16x4_f32(false, a, false, b3, (short)0, acc3, false, false);
  }

  __syncthreads();  // LDS zero-fill visible; waves done with accumulators

  // merge the 4 per-wave partial tiles via LDS float atomics (ds_add_f32).
  // C/D layout: VGPR j -> row (j + 8h), col = m within each 16-wide N-tile.
  #pragma unroll
  for (int j = 0; j < 8; ++j) {
    float* dst = &tile_lds[(j + 8 * h) * EMB_DIM + m];
    atomicAdd(dst + 0,  acc0[j]);
    atomicAdd(dst + 16, acc1[j]);
    atomicAdd(dst + 32, acc2[j]);
    atomicAdd(dst + 48, acc3[j]);
  }

  __syncthreads();

  // coalesced store of the finished 16x64 tile
  float4* out4 = (float4*)(out + (long long)base_row * EMB_DIM);
  const float4* lds4 = (const float4*)tile_lds;
  for (int i = tid; i < (TILE_M * EMB_DIM) / 4; i += BLOCK_THREADS) out4[i] = lds4[i];
}

extern "C" void kernel_launch(void* const* d_in, const int* in_sizes, int n_in,
                              void* d_out, int out_size, void* d_ws, size_t ws_size,
                              hipStream_t stream) {
  const int*   row_idx = (const int*)d_in[0];
  const int*   col_idx = (const int*)d_in[1];
  const float* data    = (const float*)d_in[2];
  // d_in[3] = num_ids (device scalar) — derived host-side from out_size instead
  const float* emb     = (const float*)d_in[4];
  float*       out     = (float*)d_out;

  const int nnz       = in_sizes[0];
  const int num_ids   = out_size / EMB_DIM;
  const int num_tiles = (num_ids + TILE_M - 1) / TILE_M;

  hipLaunchKernelGGL(sess_emb_wmma_kernel, dim3(num_tiles), dim3(BLOCK_THREADS), 0, stream,
                     row_idx, col_idx, data, emb, out, nnz);
}